// Attention_Module_9534827397762
// MI455X (gfx1250) — compile-verified
//
#include <hip/hip_runtime.h>

typedef __attribute__((ext_vector_type(2))) float v2f;
typedef __attribute__((ext_vector_type(4))) float v4f;
typedef __attribute__((ext_vector_type(8))) float v8f;

namespace {
constexpr int kN      = 32;
constexpr int kCin    = 4096;
constexpr int kHW     = 784;            // 28*28
constexpr int kCout   = 4;
constexpr int kMPad   = 16;             // A-operand rows (WMMA M)
constexpr int kCTiles = kHW / 16;       // 49 column tiles of 16 spatial positions
constexpr int kKSplit = 4;              // split Cin for parallelism (bandwidth)
constexpr int kKChunk = kCin / kKSplit; // 1024
constexpr int kWavesPerBlock = 4;
}

// ---------------------------------------------------------------------------
// K0: zero-pad fc weights (4,4096) -> (16,4096) so the WMMA A-operand can be
// loaded with no per-iteration masking (keeps VALU out of the hot loop).
// ---------------------------------------------------------------------------
__global__ __launch_bounds__(256) void pad_w_kernel(
    const float* __restrict__ w, float* __restrict__ wpad)
{
  const int i = blockIdx.x * 256 + threadIdx.x;    // 0 .. 16*4096-1
  const int m = i >> 12;                            // /4096
  wpad[i] = (m < kCout) ? w[i] : 0.f;
}

// ---------------------------------------------------------------------------
// K1: cams partial GEMM via V_WMMA_F32_16X16X4_F32.
//   D(16x16) += A(16x4) * B(4x16); A = wpad rows, B = 4 consecutive channels
//   x 16 spatial columns of x[n].
// Per ISA 7.12.2 (32-bit operands): lane = 16h+m holds K=2h,2h+1 in its two
// A/B VGPRs; C/D row M=r lives in VGPR r (lanes 0-15) and M=r+8 (lanes 16-31).
// Inner loop: 1x global_load_b64 (A, cache-hot) + 2x global_load_b32 NT (B)
// + 1x v_wmma — no VALU in the dependency chain.
// ---------------------------------------------------------------------------
__global__ __launch_bounds__(128) void cam_gemm_kernel(
    const float* __restrict__ x, const float* __restrict__ wpad,
    float* __restrict__ cpart)
{
  const int lane = threadIdx.x & 31;
  const int wid  = threadIdx.x >> 5;
  const int m    = lane & 15;   // A row / B,D column within tile
  const int h    = lane >> 4;   // half-wave: K offset 2h

  const int n     = blockIdx.x;                        // image
  const int kpart = blockIdx.y;                        // K split
  const int t     = blockIdx.z * kWavesPerBlock + wid; // spatial column tile
  if (t >= kCTiles) return;                            // wave-uniform exit
  const int s = t * 16 + m;                            // spatial column

  const float* xn   = x + (size_t)n * kCin * kHW;
  const float* wrow = wpad + m * kCin + 2 * h;
  const float* xcol = xn + 2 * h * kHW + s;
  const int k0 = kpart * kKChunk;

  v8f c = {0.f, 0.f, 0.f, 0.f, 0.f, 0.f, 0.f, 0.f};
#pragma unroll 8
  for (int k = k0; k < k0 + kKChunk; k += 4) {
    // A operand: two consecutive padded-w values (8B-aligned b64, L2-hot)
    v2f a = *(const v2f*)(wrow + k);
    // B operand: channels k+2h, k+2h+1 at column s (coalesced 64B/half-wave)
    v2f b;
    b.x = __builtin_nontemporal_load(xcol + k * kHW);
    b.y = __builtin_nontemporal_load(xcol + k * kHW + kHW);
    c = __builtin_amdgcn_wmma_f32_16x16x4_f32(false, a, false, b, (short)0, c,
                                              false, false);
  }

  // D rows 0..3 (= o) live in c[0..3] on lanes 0-15; rows 4..15 are zero.
  if (h == 0) {
    float* dst = cpart + (((size_t)kpart * kN + n) * kCout) * kHW + s;
    dst[0 * kHW] = c[0];
    dst[1 * kHW] = c[1];
    dst[2 * kHW] = c[2];
    dst[3 * kHW] = c[3];
  }
}

// ---------------------------------------------------------------------------
// K2: one block per image n. Sum K-partials, ReLU, per-o max (int-monotone
// trick on non-negative floats), threshold by gama, mean of kept cams.
// ---------------------------------------------------------------------------
__global__ __launch_bounds__(256) void drop_kernel(
    const float* __restrict__ cpart, const float* __restrict__ gama_p,
    float* __restrict__ mdrop)
{
  __shared__ float lcam[kCout][kHW];
  __shared__ int   lmax[kCout];
  const int n   = blockIdx.x;
  const int tid = threadIdx.x;
  if (tid < kCout) lmax[tid] = 0;
  __syncthreads();

  float tmax[kCout] = {0.f, 0.f, 0.f, 0.f};
  for (int s = tid; s < kHW; s += 256) {
#pragma unroll
    for (int o = 0; o < kCout; ++o) {
      float v = 0.f;
#pragma unroll
      for (int kp = 0; kp < kKSplit; ++kp)
        v += cpart[(((size_t)kp * kN + n) * kCout + o) * kHW + s];
      v = fmaxf(v, 0.f);            // relu
      lcam[o][s] = v;
      tmax[o] = fmaxf(tmax[o], v);
    }
  }
#pragma unroll
  for (int o = 0; o < kCout; ++o)
    atomicMax(&lmax[o], __float_as_int(tmax[o]));   // all values >= 0
  __syncthreads();

  const float g = *gama_p;
  float thr[kCout];
#pragma unroll
  for (int o = 0; o < kCout; ++o) thr[o] = g * __int_as_float(lmax[o]);

  for (int s = tid; s < kHW; s += 256) {
    float sum = 0.f;
#pragma unroll
    for (int o = 0; o < kCout; ++o) {
      float v = lcam[o][s];
      sum += (v <= thr[o]) ? v : 0.f;   // drop where cams > thr
    }
    mdrop[n * kHW + s] = 0.25f * sum;
  }
}

// ---------------------------------------------------------------------------
// K3: out = x * mdrop, float4-vectorized streaming pass (non-temporal).
// ---------------------------------------------------------------------------
__global__ __launch_bounds__(256) void scale_kernel(
    const v4f* __restrict__ x4, const v4f* __restrict__ md4,
    v4f* __restrict__ out4)
{
  constexpr int kQ     = kHW / 4;            // 196 float4 per (n,c)
  constexpr int kTotal = kN * kCin * kQ;     // 25,690,112
  const int id = blockIdx.x * 256 + threadIdx.x;
  if (id >= kTotal) return;
  const int nc = id / kQ;
  const int sq = id - nc * kQ;
  const int n  = nc >> 12;                   // / 4096

  v4f xv = __builtin_nontemporal_load(x4 + id);
  v4f mv = md4[n * kQ + sq];
  __builtin_nontemporal_store(xv * mv, out4 + id);
}

// ---------------------------------------------------------------------------
extern "C" void kernel_launch(void* const* d_in, const int* in_sizes, int n_in,
                              void* d_out, int out_size, void* d_ws, size_t ws_size,
                              hipStream_t stream) {
  (void)in_sizes; (void)n_in; (void)out_size; (void)ws_size;
  const float* x    = (const float*)d_in[0];   // (32,4096,28,28) f32
  const float* w    = (const float*)d_in[1];   // (4,4096,1,1)    f32
  const float* gama = (const float*)d_in[2];   // scalar f32 on device

  // Workspace: wpad [16][4096] | cpart [kKSplit][N][Cout][HW] | mdrop [N][HW]
  float* wpad  = (float*)d_ws;                                   // 65536 f
  float* cpart = wpad + (size_t)kMPad * kCin;                    // 401408 f
  float* mdrop = cpart + (size_t)kKSplit * kN * kCout * kHW;     // 25088 f

  pad_w_kernel<<<(kMPad * kCin) / 256, 256, 0, stream>>>(w, wpad);

  dim3 g1(kN, kKSplit, (kCTiles + kWavesPerBlock - 1) / kWavesPerBlock);
  cam_gemm_kernel<<<g1, kWavesPerBlock * 32, 0, stream>>>(x, wpad, cpart);

  drop_kernel<<<kN, 256, 0, stream>>>(cpart, gama, mdrop);

  constexpr int kTotal4 = kN * kCin * (kHW / 4);
  scale_kernel<<<(kTotal4 + 255) / 256, 256, 0, stream>>>(
      (const v4f*)x, (const v4f*)mdrop, (v4f*)d_out);
}